// Reasoning_65730179498619
// MI455X (gfx1250) — compile-verified
//
#include <hip/hip_runtime.h>
#include <math.h>

// ---------------------------------------------------------------------------
// Problem sizes (fixed by reference): B=256, C=512, HW=64,
// K0=16384, K1=4096, NC=1000.
// d_in order: fs0, fs1, W0, b0, W1, b1, xi0, xi1, y
// d_out: y_ [256*1000] floats, then loss [1] float.
// ---------------------------------------------------------------------------

typedef __attribute__((ext_vector_type(2))) float v2f;
typedef __attribute__((ext_vector_type(8))) float v8f;

#define BATCH 256
#define CHN   512
#define K0    16384
#define K1    4096
#define NC    1000

#define BM 128
#define BN 64
#define BK 32
#define LDT 36   // padded LDS row stride (floats): 16B-aligned, conflict-free

static __device__ __forceinline__ v8f wmma_f32_16x16x4(v2f a, v2f b, v8f c) {
#if defined(__gfx1250__) && __has_builtin(__builtin_amdgcn_wmma_f32_16x16x4_f32)
  // (neg_a, A, neg_b, B, c_mod, C, reuse_a, reuse_b)
  return __builtin_amdgcn_wmma_f32_16x16x4_f32(false, a, false, b, (short)0, c,
                                               false, false);
#else
  // Host pass / non-gfx1250 placeholder (never executed on target device).
  c[0] += a[0] * b[0];
  return c;
#endif
}

static __device__ __forceinline__ float sigmoidf_(float x) {
  return 1.0f / (1.0f + __expf(-x));
}

// ---------------------------------------------------------------------------
// Zero f0/f1 region of workspace.
// ---------------------------------------------------------------------------
__global__ void k_zero(float4* __restrict__ p, int n4) {
  int i = blockIdx.x * blockDim.x + threadIdx.x;
  if (i < n4) p[i] = float4{0.f, 0.f, 0.f, 0.f};
}

// ---------------------------------------------------------------------------
// Adaptive-avg-pool 8x8 -> 1: one wave per (b,c) entry, for fs0 and fs1.
// ---------------------------------------------------------------------------
__global__ void k_pool(const float* __restrict__ fs0,
                       const float* __restrict__ fs1,
                       float* __restrict__ t0, float* __restrict__ t1) {
  int wid = (blockIdx.x * blockDim.x + threadIdx.x) >> 5;  // global wave id
  int lane = threadIdx.x & 31;
  if (wid >= BATCH * CHN) return;

  const float2* p0 = (const float2*)(fs0 + (size_t)wid * 64);
  float2 a = p0[lane];
  float s0 = a.x + a.y;
  const float2* p1 = (const float2*)(fs1 + (size_t)wid * 64);
  float2 b = p1[lane];
  float s1 = b.x + b.y;
#pragma unroll
  for (int o = 16; o > 0; o >>= 1) {
    s0 += __shfl_xor(s0, o, 32);
    s1 += __shfl_xor(s1, o, 32);
  }
  if (lane == 0) {
    t0[wid] = s0 * (1.0f / 64.0f);
    t1[wid] = s1 * (1.0f / 64.0f);
  }
}

// ---------------------------------------------------------------------------
// Deterministic last-wins scatter: one thread per batch row, serial over C.
// ---------------------------------------------------------------------------
__global__ void k_scatter(const int* __restrict__ xi0,
                          const int* __restrict__ xi1,
                          const float* __restrict__ t0,
                          const float* __restrict__ t1,
                          float* __restrict__ f0, float* __restrict__ f1) {
  int b = blockIdx.x * blockDim.x + threadIdx.x;
  if (b >= BATCH) return;
  for (int c = 0; c < CHN; ++c)
    f0[(size_t)b * K0 + xi0[b * CHN + c]] = t0[b * CHN + c];
  for (int c = 0; c < CHN; ++c)
    f1[(size_t)b * K1 + xi1[b * CHN + c]] = t1[b * CHN + c];
}

// ---------------------------------------------------------------------------
// WMMA GEMM: C[m,n] = sigmoid( sum_k A[m,k]*W[n,k] + bias[n] ), n < Nmax.
// Block: 256 threads (8 waves), tile BM x BN, K stepped by BK (double buffer).
// Wave grid 4(M) x 2(N); per wave 2x2 subtiles of 16x16 (v8f accumulators).
// ---------------------------------------------------------------------------
__global__ __launch_bounds__(256) void k_gemm_sig(
    const float* __restrict__ A, const float* __restrict__ W,
    const float* __restrict__ bias, float* __restrict__ C,
    int K, int Nld, int Nmax) {
  __shared__ float sA[2][BM * LDT];
  __shared__ float sB[2][BN * LDT];

  const int tid = threadIdx.x;
  const int m0 = blockIdx.y * BM;
  const int n0 = blockIdx.x * BN;
  const int lane = tid & 31;
  const int wid = tid >> 5;
  const int wm = wid & 3;       // 0..3 -> 32-row slab
  const int wn = wid >> 2;      // 0..1 -> 32-col slab
  const int half = lane >> 4;   // K-pair select (ISA f32 WMMA layout)
  const int ln = lane & 15;     // M (A frag) / N (B frag) within subtile

  v8f acc[2][2];
#pragma unroll
  for (int i = 0; i < 2; ++i)
#pragma unroll
    for (int j = 0; j < 2; ++j) acc[i][j] = (v8f){0, 0, 0, 0, 0, 0, 0, 0};

  const int Ktiles = K / BK;

  auto load_tiles = [&](int buf, int kt) {
    const int k0 = kt * BK;
    // A tile: BM x BK  (128*32 floats = 1024 float4, 4 per thread)
#pragma unroll
    for (int i = 0; i < 4; ++i) {
      int q = tid + i * 256;
      int r = q >> 3;
      int cc = (q & 7) * 4;
      float4 v = *(const float4*)(A + (size_t)(m0 + r) * K + k0 + cc);
      *(float4*)(&sA[buf][r * LDT + cc]) = v;
    }
    // W tile: BN x BK (row-major [N][K]), guarded by Nmax
#pragma unroll
    for (int i = 0; i < 2; ++i) {
      int q = tid + i * 256;
      int r = q >> 3;
      int cc = (q & 7) * 4;
      int gn = n0 + r;
      float4 v = float4{0.f, 0.f, 0.f, 0.f};
      if (gn < Nmax) {
        v = *(const float4*)(W + (size_t)gn * K + k0 + cc);
        // stream-ahead prefetch of the W0/W1 stream (global_prefetch_b8)
        if (k0 + 2 * BK < K)
          __builtin_prefetch(W + (size_t)gn * K + k0 + 2 * BK + cc, 0, 1);
      }
      *(float4*)(&sB[buf][r * LDT + cc]) = v;
    }
  };

  load_tiles(0, 0);
  __syncthreads();

  for (int kt = 0; kt < Ktiles; ++kt) {
    const int cur = kt & 1;
    if (kt + 1 < Ktiles) load_tiles(cur ^ 1, kt + 1);

#pragma unroll
    for (int kk = 0; kk < BK; kk += 4) {
      v2f af[2], bf[2];
#pragma unroll
      for (int i = 0; i < 2; ++i)
        af[i] = *(const v2f*)&sA[cur][(32 * wm + 16 * i + ln) * LDT + kk + 2 * half];
#pragma unroll
      for (int j = 0; j < 2; ++j)
        bf[j] = *(const v2f*)&sB[cur][(32 * wn + 16 * j + ln) * LDT + kk + 2 * half];
#pragma unroll
      for (int i = 0; i < 2; ++i)
#pragma unroll
        for (int j = 0; j < 2; ++j)
          acc[i][j] = wmma_f32_16x16x4(af[i], bf[j], acc[i][j]);
    }
    __syncthreads();
  }

  // Epilogue: C layout -> VGPR v holds (M = v + 8*half, N = ln) of subtile.
#pragma unroll
  for (int j = 0; j < 2; ++j) {
    int col = n0 + 32 * wn + 16 * j + ln;
    if (col < Nmax) {
      float bv = bias[col];
#pragma unroll
      for (int i = 0; i < 2; ++i) {
#pragma unroll
        for (int v = 0; v < 8; ++v) {
          int row = m0 + 32 * wm + 16 * i + 8 * half + v;
          float x = acc[i][j][v] + bv;
          C[(size_t)row * Nld + col] = sigmoidf_(x);
        }
      }
    }
  }
}

// ---------------------------------------------------------------------------
// BCE-with-logits partial sums: sum(softplus(h) - h*f1), 4096 elems per block.
// ---------------------------------------------------------------------------
__global__ void k_be(const float* __restrict__ h, const float* __restrict__ f1,
                     float* __restrict__ part) {
  __shared__ float red[256];
  int tid = threadIdx.x;
  size_t base = (size_t)blockIdx.x * 4096;
  float s = 0.f;
#pragma unroll
  for (int k = 0; k < 16; ++k) {
    size_t i = base + tid + k * 256;
    float x = h[i];
    float sp = x > 0.f ? x + log1pf(__expf(-x)) : log1pf(__expf(x));
    s += sp - x * f1[i];
  }
  red[tid] = s;
  __syncthreads();
  for (int o = 128; o > 0; o >>= 1) {
    if (tid < o) red[tid] += red[tid + o];
    __syncthreads();
  }
  if (tid == 0) part[blockIdx.x] = red[0];
}

// ---------------------------------------------------------------------------
// Per-row cross-entropy term: logsumexp(y_[b,:]) - y_[b, y[b]].
// ---------------------------------------------------------------------------
__global__ void k_ce(const float* __restrict__ yhat, const int* __restrict__ y,
                     float* __restrict__ cerow) {
  __shared__ float red[256];
  int b = blockIdx.x;
  int tid = threadIdx.x;
  const float* row = yhat + (size_t)b * NC;

  float m = -1e30f;
  for (int j = tid; j < NC; j += 256) m = fmaxf(m, row[j]);
  red[tid] = m;
  __syncthreads();
  for (int o = 128; o > 0; o >>= 1) {
    if (tid < o) red[tid] = fmaxf(red[tid], red[tid + o]);
    __syncthreads();
  }
  float mx = red[0];
  __syncthreads();

  float s = 0.f;
  for (int j = tid; j < NC; j += 256) s += __expf(row[j] - mx);
  red[tid] = s;
  __syncthreads();
  for (int o = 128; o > 0; o >>= 1) {
    if (tid < o) red[tid] += red[tid + o];
    __syncthreads();
  }
  if (tid == 0) cerow[b] = mx + logf(red[0]) - row[y[b]];
}

// ---------------------------------------------------------------------------
// Combine: loss = mean(ce) + mean(be).
// ---------------------------------------------------------------------------
__global__ void k_final(const float* __restrict__ part,
                        const float* __restrict__ cerow,
                        float* __restrict__ loss) {
  __shared__ float r1[256], r2[256];
  int tid = threadIdx.x;
  r1[tid] = part[tid];
  r2[tid] = cerow[tid];
  __syncthreads();
  for (int o = 128; o > 0; o >>= 1) {
    if (tid < o) {
      r1[tid] += r1[tid + o];
      r2[tid] += r2[tid + o];
    }
    __syncthreads();
  }
  if (tid == 0)
    *loss = r2[0] * (1.0f / BATCH) + r1[0] * (1.0f / ((float)BATCH * K1));
}

// ---------------------------------------------------------------------------
extern "C" void kernel_launch(void* const* d_in, const int* in_sizes, int n_in,
                              void* d_out, int out_size, void* d_ws,
                              size_t ws_size, hipStream_t stream) {
  (void)in_sizes; (void)n_in; (void)out_size; (void)ws_size;
  const float* fs0 = (const float*)d_in[0];
  const float* fs1 = (const float*)d_in[1];
  const float* W0  = (const float*)d_in[2];
  const float* b0  = (const float*)d_in[3];
  const float* W1  = (const float*)d_in[4];
  const float* b1  = (const float*)d_in[5];
  const int*   xi0 = (const int*)d_in[6];
  const int*   xi1 = (const int*)d_in[7];
  const int*   y   = (const int*)d_in[8];
  float* out = (float*)d_out;
  float* ws  = (float*)d_ws;

  // Workspace layout (floats); total ~26.3 MB.
  float* t0 = ws;                       // 131072
  float* t1 = t0 + BATCH * CHN;         // 131072
  float* f0 = t1 + BATCH * CHN;         // 4194304
  float* f1 = f0 + (size_t)BATCH * K0;  // 1048576
  float* h  = f1 + (size_t)BATCH * K1;  // 1048576
  float* bepart = h + (size_t)BATCH * K1;  // 256
  float* cerow  = bepart + 256;            // 256

  // 1) zero the scatter targets (f0 + f1, contiguous: 5,242,880 floats)
  k_zero<<<5120, 256, 0, stream>>>((float4*)f0, 5242880 / 4);
  // 2) 8x8 average pool (one wave per (b,c))
  k_pool<<<16384, 256, 0, stream>>>(fs0, fs1, t0, t1);
  // 3) deterministic last-wins scatter into dense f0/f1
  k_scatter<<<1, 256, 0, stream>>>(xi0, xi1, t0, t1, f0, f1);
  // 4) h = sigmoid(f0 @ W0^T + b0): 256 x 4096, K=16384
  k_gemm_sig<<<dim3(K1 / BN, BATCH / BM), 256, 0, stream>>>(f0, W0, b0, h,
                                                            K0, K1, K1);
  // 5) y_ = sigmoid(h @ W1^T + b1): 256 x 1000, K=4096  (16 N-tiles cover 1024)
  k_gemm_sig<<<dim3(16, BATCH / BM), 256, 0, stream>>>(h, W1, b1, out,
                                                       K1, NC, NC);
  // 6) be partials over 256*4096 elements
  k_be<<<256, 256, 0, stream>>>(h, f1, bepart);
  // 7) per-row CE term on y_
  k_ce<<<256, 256, 0, stream>>>(out, y, cerow);
  // 8) loss = mean(ce) + mean(be)
  k_final<<<1, 256, 0, stream>>>(bepart, cerow, out + (size_t)BATCH * NC);
}